// DGEmbedding_31129922961736
// MI455X (gfx1250) — compile-verified
//
#include <hip/hip_runtime.h>
#include <math.h>
#include <stdint.h>

#define NSPH 7
#define NRAD 6
#define LK 42          // NSPH * NRAD
#define BLOCK 256
#define WPB 8          // waves per block (wave32)

#if defined(__has_builtin)
#if __has_builtin(__builtin_amdgcn_global_load_async_to_lds_b64)
#define HAVE_ASYNC_LDS 1
#endif
#endif

#if defined(HAVE_ASYNC_LDS)
// 64-bit async payload type, per the builtin's actual signature:
// (v2i32 AS1*, v2i32 AS3*, imm offset, imm cpol)
typedef int v2i __attribute__((vector_size(2 * sizeof(int))));
typedef __attribute__((address_space(1))) v2i* v2i_g;
typedef __attribute__((address_space(3))) v2i* v2i_l;
#endif

// ---------------------------------------------------------------------------
// Kernel A: per-edge radial basis. One thread per edge; each wave stages its
// 32x42 rbf tile (+32x6 dist-emb tile) in LDS, then writes fully coalesced.
// rbf is written with default (RT) temporal hint so it stays L2-resident for
// the gather kernel; dist_embs (final output) is written non-temporally.
// ---------------------------------------------------------------------------
__global__ __launch_bounds__(BLOCK) void dg_edge_kernel(
    const float* __restrict__ dists, const float* __restrict__ freq,
    float* __restrict__ demb, float* __restrict__ rbf, int E)
{
#pragma clang fp contract(off)
  __shared__ __align__(16) float rbf_s[WPB][32 * LK];
  __shared__ __align__(16) float demb_s[WPB][32 * NRAD];

  const int tid  = threadIdx.x;
  const int lane = tid & 31;
  const int wv   = tid >> 5;
  const int i     = blockIdx.x * BLOCK + tid;
  const int ebase = blockIdx.x * BLOCK + wv * 32;

  float d = 1.0f;
  if (i < E) d = __builtin_nontemporal_load(&dists[i]);
  const float dn = d * 0.2f;  // dists / CUTOFF

  // envelope(x) = 1/x + a x^5 + b x^6 + c x^7, p=6: a=-28, b=48, c=21
  const float x2 = dn * dn;
  const float p5 = (x2 * x2) * dn;  // integer_pow(x,5) like XLA
  const float env = 1.0f / dn + (-28.0f * p5) + ((48.0f * p5) * dn)
                  + (((21.0f * p5) * dn) * dn);

  // dist embeddings: env * sin(freq_k * dn)
  for (int k = 0; k < NRAD; ++k) {
    float s = sinf(freq[k] * dn);
    demb_s[wv][lane * NRAD + k] = env * s;
  }

  // spherical Bessel basis, exact reference recurrence order
  float* row = &rbf_s[wv][lane * LK];
  for (int l = 0; l < NSPH; ++l) {
    for (int k = 0; k < NRAD; ++k) {
      const float z = (float)(((double)(k + 1) + 0.5 * (double)l) *
                              3.14159265358979323846);
      const float x = dn * z;
      float s, c;
      sincosf(x, &s, &c);
      float val = s / x;  // j0
      if (l > 0) {
        float jm = val;
        float jc = s / (x * x) - c / x;  // j1
        for (int m = 1; m < l; ++m) {
          float t = (2.0f * (float)m + 1.0f) / x * jc - jm;
          jm = jc; jc = t;
        }
        val = jc;
      }
      row[l * NRAD + k] = env * val;
    }
  }

  __syncthreads();

  // coalesced rbf writes (regular temporal -> L2-resident for gather)
  {
    const size_t base = (size_t)ebase * LK;
    const size_t lim  = (size_t)E * LK;
    for (int it = 0; it < LK; ++it) {
      const int idx = it * 32 + lane;
      const size_t a = base + (size_t)idx;
      if (a < lim) rbf[a] = rbf_s[wv][idx];
    }
  }
  // coalesced non-temporal dist-emb writes (streaming final output)
  {
    const size_t base = (size_t)ebase * NRAD;
    const size_t lim  = (size_t)E * NRAD;
    for (int it = 0; it < NRAD; ++it) {
      const int idx = it * 32 + lane;
      const size_t a = base + (size_t)idx;
      if (a < lim) __builtin_nontemporal_store(demb_s[wv][idx], &demb[a]);
    }
  }
}

// ---------------------------------------------------------------------------
// Kernel B: per-triplet combine. Each wave owns 32 triplets: Legendre stack
// into LDS, async-gather 32 rbf rows (168 B each) into the wave's LDS tile
// via global_load_async_to_lds_b64 (ASYNCcnt), then 42 fully coalesced
// multiply + non-temporal-store iterations.
// ---------------------------------------------------------------------------
__global__ __launch_bounds__(BLOCK) void dg_triplet_kernel(
    const float* __restrict__ angles, const int* __restrict__ kj,
    const float* __restrict__ rbf, float* __restrict__ out, int T)
{
#pragma clang fp contract(off)
  __shared__ __align__(16) float rows_s[WPB][32 * LK];
  __shared__ __align__(16) float cbf_s[WPB][32][8];

  const int tid  = threadIdx.x;
  const int lane = tid & 31;
  const int wv   = tid >> 5;
  const int t     = blockIdx.x * BLOCK + tid;
  const int tbase = blockIdx.x * BLOCK + wv * 32;

  float ang = 0.0f; int e = 0;
  if (t < T) {
    ang = __builtin_nontemporal_load(&angles[t]);
    e   = __builtin_nontemporal_load(&kj[t]);
  }
  const float c = cosf(ang);

  // Legendre P_0..P_6(c), reference recurrence
  {
    float pm = 1.0f, pc = c;
    cbf_s[wv][lane][0] = pm;
    cbf_s[wv][lane][1] = pc;
    for (int l = 1; l < NSPH - 1; ++l) {
      float pn = ((2.0f * (float)l + 1.0f) * c * pc - (float)l * pm) /
                 (float)(l + 1);
      pm = pc; pc = pn;
      cbf_s[wv][lane][l + 1] = pn;
    }
  }

  const float* rowp = rbf + (size_t)e * LK;   // gathered row (168 B, 8B-aligned)
  float* lrow = &rows_s[wv][lane * LK];

#if defined(HAVE_ASYNC_LDS)
#pragma unroll
  for (int j = 0; j < LK / 2; ++j) {
    __builtin_amdgcn_global_load_async_to_lds_b64(
        (v2i_g)(uintptr_t)(rowp + 2 * j),
        (v2i_l)(uint32_t)(uintptr_t)(lrow + 2 * j),
        0, 0);
  }
#else
  for (int j = 0; j < LK / 2; ++j) {
    float2 v = *(const float2*)(rowp + 2 * j);
    *(float2*)(lrow + 2 * j) = v;
  }
#endif

  __syncthreads();              // cbf_s visible across the wave
#if defined(HAVE_ASYNC_LDS)
  asm volatile("s_wait_asynccnt 0" ::: "memory");  // wave's async gather done
#endif

  const size_t obase = (size_t)tbase * LK;
  for (int it = 0; it < LK; ++it) {
    const int idx = it * 32 + lane;       // linear within 32x42 wave tile
    const int tt  = idx / LK;             // local triplet
    const int j   = idx - tt * LK;        // column 0..41
    const int l   = j / NRAD;             // spherical order
    const float v = rows_s[wv][idx] * cbf_s[wv][tt][l];
    if (tbase + tt < T)
      __builtin_nontemporal_store(v, &out[obase + (size_t)idx]);
  }
}

// ---------------------------------------------------------------------------
extern "C" void kernel_launch(void* const* d_in, const int* in_sizes, int n_in,
                              void* d_out, int out_size, void* d_ws, size_t ws_size,
                              hipStream_t stream) {
  const float* dists  = (const float*)d_in[0];
  const float* angles = (const float*)d_in[1];
  const int*   kj     = (const int*)d_in[2];
  const float* freq   = (const float*)d_in[3];
  const int E = in_sizes[0];
  const int T = in_sizes[1];
  (void)n_in; (void)out_size; (void)ws_size;

  float* demb    = (float*)d_out;                      // [E,6]
  float* out_big = (float*)d_out + (size_t)E * NRAD;   // [T,42]
  float* rbf     = (float*)d_ws;                       // [E,42] scratch

  const int gE = (E + BLOCK - 1) / BLOCK;
  const int gT = (T + BLOCK - 1) / BLOCK;
  dg_edge_kernel<<<gE, BLOCK, 0, stream>>>(dists, freq, demb, rbf, E);
  dg_triplet_kernel<<<gT, BLOCK, 0, stream>>>(angles, kj, rbf, out_big, T);
}